// GSDN_32744830664752
// MI455X (gfx1250) — compile-verified
//
#include <hip/hip_runtime.h>
#include <math.h>

typedef __attribute__((ext_vector_type(2))) float v2f;
typedef __attribute__((ext_vector_type(4))) float v4f;
typedef __attribute__((ext_vector_type(8))) float v8f;

#define CH 192
#define HW 65536
#define TILE_PX 64
#define LDS_STRIDE 72   /* words: 288B rows -> 16B aligned, bank-clean K-loads */

#define PEDESTAL_F   1.4551915228366852e-11f  /* 2^-36 */
#define GAMMA_BOUND_F 3.814697265625e-06f     /* 2^-18 */
#define BETA_BOUND_F  1.0000072759132523e-03f /* sqrt(1e-6 + 2^-36) */

__device__ __forceinline__ float reparam_g(float v) {
    float m = fmaxf(v, GAMMA_BOUND_F);
    return m * m - PEDESTAL_F;
}
__device__ __forceinline__ float reparam_b(float v) {
    float m = fmaxf(v, BETA_BOUND_F);
    return m * m - PEDESTAL_F;
}

__global__ __launch_bounds__(128)
void gsdn_fused(const float* __restrict__ x,
                const float* __restrict__ beta,
                const float* __restrict__ gamma,
                const float* __restrict__ beta2,
                const float* __restrict__ gamma2,
                float* __restrict__ out)
{
    __shared__ float buf0[CH * LDS_STRIDE]; /* Xin, later Xc^2 */
    __shared__ float buf1[CH * LDS_STRIDE]; /* Xc */

    const int tid  = threadIdx.x;
    const int lane = tid & 31;
    const int wave = tid >> 5;
    const int nlo  = lane & 15;   /* B/C/D column within 16-px subtile; A row in block */
    const int hi   = lane >> 4;   /* lane half */

    const int q  = blockIdx.x * TILE_PX;   /* global pixel id over B*H*W */
    const int bb = q >> 16;                /* batch (HW = 65536) */
    const int j0 = q & (HW - 1);
    const float* xin = x   + (size_t)bb * CH * HW + j0;
    float*       op  = out + (size_t)bb * CH * HW + j0;

    /* ---- Phase A: async copy of 192x64 input tile direct to LDS ----
       GLOBAL_LOAD_ASYNC_TO_LDS_B128, GVS mode: saddr = tile base,
       vaddr = per-lane byte offset, vdst = LDS byte offset.          */
    #pragma unroll
    for (int i = 0; i < 24; ++i) {
        int idx = tid + i * 128;          /* float4 index over [192][16] */
        int row = idx >> 4;
        int c4  = (idx & 15) << 2;
        unsigned goff    = (unsigned)((row * HW + c4) * 4);
        unsigned lds_off = (unsigned)(uintptr_t)&buf0[row * LDS_STRIDE + c4];
        asm volatile("global_load_async_to_lds_b128 %0, %1, %2"
                     :
                     : "v"(lds_off), "v"(goff), "s"(xin)
                     : "memory");
    }
    asm volatile("s_wait_asynccnt 0x0" ::: "memory");
    __syncthreads();

    const int koff = hi << 1;

    /* ---- Phase B: mean = reparam(gamma2) * Xin ; Xc = Xin - mean - b2 ---- */
    for (int mi = 0; mi < 3; ++mi) {
        const int mBase = ((wave * 3 + mi) << 4);
        const int arow  = mBase + nlo;

        v8f acc[4];
        #pragma unroll
        for (int p = 0; p < 4; ++p) acc[p] = (v8f){0.f,0.f,0.f,0.f,0.f,0.f,0.f,0.f};

        #pragma unroll 4
        for (int kk = 0; kk < 48; ++kk) {
            const int k = (kk << 2) + koff;
            v2f a = *(const v2f*)(gamma2 + arow * CH + k);
            a.x = reparam_g(a.x);
            a.y = reparam_g(a.y);
            const float* r0 = &buf0[k * LDS_STRIDE + nlo];
            const float* r1 = r0 + LDS_STRIDE;
            v2f b0 = { r0[0],  r1[0]  };
            v2f b1 = { r0[16], r1[16] };
            v2f b2 = { r0[32], r1[32] };
            v2f b3 = { r0[48], r1[48] };
            acc[0] = __builtin_amdgcn_wmma_f32_16x16x4_f32(false, a, false, b0, (short)0, acc[0], false, false);
            acc[1] = __builtin_amdgcn_wmma_f32_16x16x4_f32(false, a, false, b1, (short)0, acc[1], false, false);
            acc[2] = __builtin_amdgcn_wmma_f32_16x16x4_f32(false, a, false, b2, (short)0, acc[2], false, false);
            acc[3] = __builtin_amdgcn_wmma_f32_16x16x4_f32(false, a, false, b3, (short)0, acc[3], false, false);
        }

        float b2r[8];
        #pragma unroll
        for (int v = 0; v < 8; ++v)
            b2r[v] = reparam_b(beta2[mBase + v + (hi << 3)]);

        #pragma unroll
        for (int p = 0; p < 4; ++p) {
            #pragma unroll
            for (int v = 0; v < 8; ++v) {
                int o = mBase + v + (hi << 3);
                int j = (p << 4) + nlo;
                float xc = buf0[o * LDS_STRIDE + j] - (acc[p][v] + b2r[v]);
                buf1[o * LDS_STRIDE + j] = xc;
            }
        }
    }
    __syncthreads();

    /* ---- Phase C: buf0 = Xc^2 ---- */
    #pragma unroll 8
    for (int i = 0; i < 96; ++i) {
        int idx = tid + i * 128;          /* over [192][64] */
        int row = idx >> 6;
        int col = idx & 63;
        float v = buf1[row * LDS_STRIDE + col];
        buf0[row * LDS_STRIDE + col] = v * v;
    }
    __syncthreads();

    /* ---- Phase D: var = reparam(gamma) * Xc^2 ; out = Xc / sqrt(var + b1) ---- */
    for (int mi = 0; mi < 3; ++mi) {
        const int mBase = ((wave * 3 + mi) << 4);
        const int arow  = mBase + nlo;

        v8f acc[4];
        #pragma unroll
        for (int p = 0; p < 4; ++p) acc[p] = (v8f){0.f,0.f,0.f,0.f,0.f,0.f,0.f,0.f};

        #pragma unroll 4
        for (int kk = 0; kk < 48; ++kk) {
            const int k = (kk << 2) + koff;
            v2f a = *(const v2f*)(gamma + arow * CH + k);
            a.x = reparam_g(a.x);
            a.y = reparam_g(a.y);
            const float* r0 = &buf0[k * LDS_STRIDE + nlo];
            const float* r1 = r0 + LDS_STRIDE;
            v2f b0 = { r0[0],  r1[0]  };
            v2f b1 = { r0[16], r1[16] };
            v2f b2 = { r0[32], r1[32] };
            v2f b3 = { r0[48], r1[48] };
            acc[0] = __builtin_amdgcn_wmma_f32_16x16x4_f32(false, a, false, b0, (short)0, acc[0], false, false);
            acc[1] = __builtin_amdgcn_wmma_f32_16x16x4_f32(false, a, false, b1, (short)0, acc[1], false, false);
            acc[2] = __builtin_amdgcn_wmma_f32_16x16x4_f32(false, a, false, b2, (short)0, acc[2], false, false);
            acc[3] = __builtin_amdgcn_wmma_f32_16x16x4_f32(false, a, false, b3, (short)0, acc[3], false, false);
        }

        float b1r[8];
        #pragma unroll
        for (int v = 0; v < 8; ++v)
            b1r[v] = reparam_b(beta[mBase + v + (hi << 3)]);

        #pragma unroll
        for (int p = 0; p < 4; ++p) {
            #pragma unroll
            for (int v = 0; v < 8; ++v) {
                int o = mBase + v + (hi << 3);
                int j = (p << 4) + nlo;
                float nrm = sqrtf(acc[p][v] + b1r[v]);
                float res = buf1[o * LDS_STRIDE + j] / nrm;
                __builtin_nontemporal_store(res, &op[(size_t)o * HW + j]);
            }
        }
    }
}

extern "C" void kernel_launch(void* const* d_in, const int* in_sizes, int n_in,
                              void* d_out, int out_size, void* d_ws, size_t ws_size,
                              hipStream_t stream) {
    (void)in_sizes; (void)n_in; (void)out_size; (void)d_ws; (void)ws_size;
    const float* x      = (const float*)d_in[0];
    const float* beta   = (const float*)d_in[1];
    const float* gamma  = (const float*)d_in[2];
    const float* beta2  = (const float*)d_in[3];
    const float* gamma2 = (const float*)d_in[4];
    float* out = (float*)d_out;

    const int total_px = 4 * HW;              /* 262144 */
    const int blocks   = total_px / TILE_PX;  /* 4096 */
    hipLaunchKernelGGL(gsdn_fused, dim3(blocks), dim3(128), 0, stream,
                       x, beta, gamma, beta2, gamma2, out);
}